// SinkhornLinear2_74371653697768
// MI455X (gfx1250) — compile-verified
//
#include <hip/hip_runtime.h>
#include <hip/hip_bf16.h>
#include <stddef.h>

typedef __attribute__((ext_vector_type(2))) float v2f;
typedef __attribute__((ext_vector_type(8))) float v8f;

#define DIM 2048          // d == o == 2048
#define MTOT 16384        // 8 * 2048 rows of x
#define TS 64             // block tile (M and N)
#define KT 32             // K tile staged in LDS
#define LDP 33            // padded LDS row stride (floats) -> conflict-free

// ---------------- Sinkhorn kernels (operate in place on W) ----------------

__global__ __launch_bounds__(256) void sk_clip(const float* __restrict__ win,
                                               float* __restrict__ w) {
    size_t i = (size_t)blockIdx.x * 256 + threadIdx.x;
    w[i] = fmaxf(win[i], 1e-8f);
}

// one block per row: reduce row sum in LDS, then divide the row by it
__global__ __launch_bounds__(256) void sk_rownorm(float* __restrict__ w) {
    __shared__ float red[256];
    const int r = blockIdx.x;
    float s = 0.0f;
    for (int c = threadIdx.x; c < DIM; c += 256)
        s += w[(size_t)r * DIM + c];
    red[threadIdx.x] = s;
    __syncthreads();
    for (int off = 128; off > 0; off >>= 1) {
        if (threadIdx.x < off) red[threadIdx.x] += red[threadIdx.x + off];
        __syncthreads();
    }
    const float rs = red[0];
    for (int c = threadIdx.x; c < DIM; c += 256)
        w[(size_t)r * DIM + c] = w[(size_t)r * DIM + c] / rs;
}

// coalesced column sums: thread owns one column, marches down rows
__global__ __launch_bounds__(256) void sk_colsum(const float* __restrict__ w,
                                                 float* __restrict__ sums) {
    const int c = blockIdx.x * 256 + threadIdx.x;
    float s = 0.0f;
    for (int r = 0; r < DIM; ++r)
        s += w[(size_t)r * DIM + c];
    sums[c] = s;
}

__global__ __launch_bounds__(256) void sk_colscale(float* __restrict__ w,
                                                   const float* __restrict__ sums) {
    const int r = blockIdx.x;
    for (int c = threadIdx.x; c < DIM; c += 256)
        w[(size_t)r * DIM + c] = w[(size_t)r * DIM + c] / sums[c];
}

// ---------------- GEMM: Y[m][n] = sum_k X[m][k] * W[n][k] ----------------
// 256 threads = 8 waves; wave (wM 0..3, wN 0..1) owns a 16x32 strip of the
// 64x64 block tile; fp32 WMMA 16x16x4 accumulation through LDS K-tiles.

__global__ __launch_bounds__(256) void gemm_xWt(const float* __restrict__ X,
                                                const float* __restrict__ W,
                                                float* __restrict__ Y) {
    __shared__ float As[TS * LDP];
    __shared__ float Bs[TS * LDP];

    const int tid   = threadIdx.x;
    const int lane  = tid & 31;
    const int wid   = tid >> 5;
    const int l15   = lane & 15;
    const int halfl = lane >> 4;        // 0 or 1
    const int waveM = wid & 3;          // 0..3 -> M offset 16*waveM
    const int waveN = wid >> 2;         // 0..1 -> N offset 32*waveN

    const int m0 = blockIdx.y * TS;     // global M tile origin
    const int n0 = blockIdx.x * TS;     // global N tile origin

    // loader mapping: thread covers rows (lrow, lrow+32), 4 floats at lcol
    const int lrow = tid >> 3;          // 0..31
    const int lcol = (tid & 7) * 4;     // 0,4,...,28

    v8f acc0 = {};
    v8f acc1 = {};

    const int aRow = 16 * waveM + l15;  // A fragment row within tile
    const int bRow0 = 32 * waveN + l15; // B fragment rows within tile
    const int bRow1 = bRow0 + 16;

    for (int k0 = 0; k0 < DIM; k0 += KT) {
        const float4 a0 = *(const float4*)(X + (size_t)(m0 + lrow)      * DIM + k0 + lcol);
        const float4 a1 = *(const float4*)(X + (size_t)(m0 + lrow + 32) * DIM + k0 + lcol);
        const float4 b0 = *(const float4*)(W + (size_t)(n0 + lrow)      * DIM + k0 + lcol);
        const float4 b1 = *(const float4*)(W + (size_t)(n0 + lrow + 32) * DIM + k0 + lcol);

        __syncthreads();   // previous iteration's LDS reads complete
        As[(lrow)      * LDP + lcol + 0] = a0.x;
        As[(lrow)      * LDP + lcol + 1] = a0.y;
        As[(lrow)      * LDP + lcol + 2] = a0.z;
        As[(lrow)      * LDP + lcol + 3] = a0.w;
        As[(lrow + 32) * LDP + lcol + 0] = a1.x;
        As[(lrow + 32) * LDP + lcol + 1] = a1.y;
        As[(lrow + 32) * LDP + lcol + 2] = a1.z;
        As[(lrow + 32) * LDP + lcol + 3] = a1.w;
        Bs[(lrow)      * LDP + lcol + 0] = b0.x;
        Bs[(lrow)      * LDP + lcol + 1] = b0.y;
        Bs[(lrow)      * LDP + lcol + 2] = b0.z;
        Bs[(lrow)      * LDP + lcol + 3] = b0.w;
        Bs[(lrow + 32) * LDP + lcol + 0] = b1.x;
        Bs[(lrow + 32) * LDP + lcol + 1] = b1.y;
        Bs[(lrow + 32) * LDP + lcol + 2] = b1.z;
        Bs[(lrow + 32) * LDP + lcol + 3] = b1.w;
        __syncthreads();

#pragma unroll
        for (int kk = 0; kk < KT; kk += 4) {
            const int kb = kk + 2 * halfl;   // per-lane K slot (ISA A/B layout)
            v2f a;
            a.x = As[aRow * LDP + kb];
            a.y = As[aRow * LDP + kb + 1];
            v2f bA;
            bA.x = Bs[bRow0 * LDP + kb];
            bA.y = Bs[bRow0 * LDP + kb + 1];
            v2f bB;
            bB.x = Bs[bRow1 * LDP + kb];
            bB.y = Bs[bRow1 * LDP + kb + 1];
            acc0 = __builtin_amdgcn_wmma_f32_16x16x4_f32(
                false, a, false, bA, (short)0, acc0, false, false);
            acc1 = __builtin_amdgcn_wmma_f32_16x16x4_f32(
                false, a, false, bB, (short)0, acc1, false, false);
        }
    }

    // C/D layout: lane L, vgpr v -> (M = v + 8*(L>>4), N = L&15)
    const int mOut = m0 + 16 * waveM + 8 * halfl;
    const int nOut = n0 + 32 * waveN + l15;
#pragma unroll
    for (int v = 0; v < 8; ++v) {
        Y[(size_t)(mOut + v) * DIM + nOut]      = acc0[v];
        Y[(size_t)(mOut + v) * DIM + nOut + 16] = acc1[v];
    }
}

// ---------------- host-side launch ----------------

extern "C" void kernel_launch(void* const* d_in, const int* in_sizes, int n_in,
                              void* d_out, int out_size, void* d_ws, size_t ws_size,
                              hipStream_t stream) {
    const float* x      = (const float*)d_in[0];   // (8, 2048, 2048) fp32
    const float* weight = (const float*)d_in[1];   // (2048, 2048) fp32

    float* y = (float*)d_out;                              // 8*2048*2048 floats
    float* w = y + (size_t)8 * 2048 * 2048;                // W output region (in-place Sinkhorn)
    float* colsums = (float*)d_ws;                         // 2048 floats scratch

    // Sinkhorn projection, in place in the W output slot
    sk_clip<<<(DIM * DIM) / 256, 256, 0, stream>>>(weight, w);
    for (int it = 0; it < 10; ++it) {
        sk_rownorm <<<DIM, 256, 0, stream>>>(w);
        sk_colsum  <<<DIM / 256, 256, 0, stream>>>(w, colsums);
        sk_colscale<<<DIM, 256, 0, stream>>>(w, colsums);
    }

    // y = x @ W^T : tiles of 64x64, grid (N/64, M/64)
    dim3 grid(DIM / TS, MTOT / TS);
    gemm_xWt<<<grid, 256, 0, stream>>>(x, w, y);
}